// NEGA_91053306675761
// MI455X (gfx1250) — compile-verified
//
#include <hip/hip_runtime.h>

typedef __attribute__((ext_vector_type(2))) float v2f;
typedef __attribute__((ext_vector_type(8))) float v8f;

#define HID 128

// ---- gfx1250 async global->LDS path (guarded: silently falls back if the
// toolchain doesn't declare the builtin under this exact name) --------------
#if defined(__has_builtin)
#if __has_builtin(__builtin_amdgcn_global_load_async_to_lds_b128)
#define USE_ASYNC_LDS 1
#endif
#endif
#ifndef USE_ASYNC_LDS
#define USE_ASYNC_LDS 0
#endif

#if USE_ASYNC_LDS
// builtin signature (from clang diagnostic): param1 = AS1 pointer to
// 'int __attribute__((vector_size(16)))', param2 = AS3 pointer to same.
typedef int vint4 __attribute__((vector_size(16)));
typedef __attribute__((address_space(1))) vint4* as1_v4i;
typedef __attribute__((address_space(3))) vint4* as3_v4i;
// flat LDS pointer's low 32 bits are the LDS offset (ISA 10.2 aperture rules),
// so an integer round-trip yields valid AS1/AS3 pointers without addrspacecast.
__device__ __forceinline__ void async_copy16(const float* g, float* l) {
    __builtin_amdgcn_global_load_async_to_lds_b128(
        (as1_v4i)(unsigned long long)(const void*)g,
        (as3_v4i)(unsigned int)(unsigned long long)(void*)l,
        0, 0);
}
#if __has_builtin(__builtin_amdgcn_s_wait_asynccnt)
#define WAIT_ASYNC() __builtin_amdgcn_s_wait_asynccnt(0)
#else
#define WAIT_ASYNC() asm volatile("s_wait_asynccnt 0x0" ::: "memory")
#endif
#endif

__device__ __forceinline__ float sigmoidf_(float x) { return 1.0f / (1.0f + __expf(-x)); }

// order-preserving float -> uint key for atomic max over floats
__device__ __forceinline__ unsigned fkey(float f) {
    int b = __float_as_int(f);
    return (unsigned)(b ^ ((b >> 31) | 0x80000000));
}
__device__ __forceinline__ float funkey(unsigned u) {
    int b = (u & 0x80000000u) ? (int)(u ^ 0x80000000u) : ~(int)u;
    return __int_as_float(b);
}

// ---------------------------------------------------------------------------
// Fused GEMM: OUT[N,128] = act( [X1 | X2][N, K1+K2] @ W[128, K1+K2]^T + bias )
// f32 WMMA 16x16x4. Block = 256 threads (8 waves). Block tile = 128 rows x 128
// cols; each wave owns a 16-col stripe and all 8 row tiles (8 x v8f acc).
// Requires K1 % 64 == 0 (or K2 == 0) so each 64-wide K chunk has one source.
// Staging uses GLOBAL_LOAD_ASYNC_TO_LDS_B128 when available.
// ---------------------------------------------------------------------------
template <int K1, int K2, int ACT>
__global__ __launch_bounds__(256) void gemm_wmma(
    const float* __restrict__ X1, const float* __restrict__ X2,
    const float* __restrict__ W, const float* __restrict__ bias,
    float* __restrict__ OUT, int nrows)
{
    constexpr int K = K1 + K2;
    __shared__ float sW[128 * 64];   // W chunk  [n][kk]
    __shared__ float sX[128 * 64];   // X chunk  [row][kk]

    const int tid  = threadIdx.x;
    const int lane = tid & 31;
    const int wave = tid >> 5;
    const int row0 = blockIdx.x * 128;

    const int ncol0 = wave * 16;
    const int lm    = lane & 15;
    const int khalf = (lane >> 4) * 2;   // 0 or 2 (ISA A/B fragment split)

    v8f acc[8];
#pragma unroll
    for (int r = 0; r < 8; ++r) acc[r] = v8f{0, 0, 0, 0, 0, 0, 0, 0};

    for (int kc = 0; kc < K; kc += 64) {
        __syncthreads();
        const float* src; int srcK, srcOff;
        if (K2 == 0 || kc < K1) { src = X1; srcK = K1; srcOff = kc; }
        else                    { src = X2; srcK = K2; srcOff = kc - K1; }
#if USE_ASYNC_LDS
        // ---- W chunk: 128 n x 64 k via async 16B/lane copies
        for (int i = tid; i < 128 * 16; i += 256) {
            const int n  = i >> 4;
            const int c4 = i & 15;
            async_copy16(W + (size_t)n * K + kc + c4 * 4, sW + (size_t)i * 4);
        }
        // ---- X chunk: 128 rows x 64 k; clamp OOB rows (their outputs are
        //      never stored, see guarded epilogue)
        for (int i = tid; i < 128 * 16; i += 256) {
            const int r  = i >> 4;
            const int c4 = i & 15;
            int grow = row0 + r;
            if (grow >= nrows) grow = nrows - 1;
            async_copy16(src + (size_t)grow * srcK + srcOff + c4 * 4,
                         sX + (size_t)i * 4);
        }
        WAIT_ASYNC();
#else
        for (int i = tid; i < 128 * 16; i += 256) {
            const int n  = i >> 4;
            const int c4 = i & 15;
            reinterpret_cast<float4*>(sW)[i] =
                reinterpret_cast<const float4*>(W + (size_t)n * K + kc)[c4];
        }
        for (int i = tid; i < 128 * 16; i += 256) {
            const int r    = i >> 4;
            const int c4   = i & 15;
            const int grow = row0 + r;
            float4 v = make_float4(0.f, 0.f, 0.f, 0.f);
            if (grow < nrows)
                v = reinterpret_cast<const float4*>(src + (size_t)grow * srcK + srcOff)[c4];
            reinterpret_cast<float4*>(sX)[i] = v;
        }
#endif
        __syncthreads();

        // ---- 16 K-steps of 4, 8 WMMAs each
#pragma unroll
        for (int ks = 0; ks < 16; ++ks) {
            const int kk = ks * 4 + khalf;
            const v2f b = *reinterpret_cast<const v2f*>(&sW[(ncol0 + lm) * 64 + kk]);
#pragma unroll
            for (int r = 0; r < 8; ++r) {
                const v2f a = *reinterpret_cast<const v2f*>(&sX[(r * 16 + lm) * 64 + kk]);
                acc[r] = __builtin_amdgcn_wmma_f32_16x16x4_f32(
                    false, a, false, b, (short)0, acc[r], false, false);
            }
        }
    }

    // ---- epilogue: D layout = VGPR i -> M = (lane<16 ? i : i+8), N = lane%16
    const float bv = bias ? bias[ncol0 + lm] : 0.0f;
#pragma unroll
    for (int r = 0; r < 8; ++r) {
#pragma unroll
        for (int i = 0; i < 8; ++i) {
            const int mrow = (lane < 16) ? i : (i + 8);
            const int grow = row0 + r * 16 + mrow;
            if (grow < nrows) {
                float v = acc[r][i] + bv;
                if (ACT == 1) v = fmaxf(v, 0.0f);
                OUT[(size_t)grow * HID + ncol0 + lm] = v;
            }
        }
    }
}

// ---------------------------------------------------------------------------
// per-node attention scores: s_src[n] = <xw[n], a_s>, s_dst[n] = <xw[n], a_d>
// ---------------------------------------------------------------------------
__global__ void node_scores(const float* __restrict__ xw,
                            const float* __restrict__ a_s,
                            const float* __restrict__ a_d,
                            float* __restrict__ s_src, float* __restrict__ s_dst, int n)
{
    const int w    = (blockIdx.x * blockDim.x + threadIdx.x) >> 5;
    const int lane = threadIdx.x & 31;
    if (w >= n) return;
    const float4 v  = reinterpret_cast<const float4*>(xw + (size_t)w * HID)[lane];
    const float4 as = reinterpret_cast<const float4*>(a_s)[lane];
    const float4 ad = reinterpret_cast<const float4*>(a_d)[lane];
    float ps = v.x * as.x + v.y * as.y + v.z * as.z + v.w * as.w;
    float pd = v.x * ad.x + v.y * ad.y + v.z * ad.z + v.w * ad.w;
    for (int off = 16; off > 0; off >>= 1) {
        ps += __shfl_xor(ps, off, 32);
        pd += __shfl_xor(pd, off, 32);
    }
    if (lane == 0) { s_src[w] = ps; s_dst[w] = pd; }
}

__global__ void fill_f32(float* __restrict__ p, float v, int count)
{
    const int i = blockIdx.x * blockDim.x + threadIdx.x;
    if (i < count) p[i] = v;
}

// e = leaky_relu(s_src[src] + s_dst[dst]); segment-max via uint-key atomics
__global__ void edge_logits(const int* __restrict__ src, const int* __restrict__ dst,
                            const float* __restrict__ s_src, const float* __restrict__ s_dst,
                            float* __restrict__ e, unsigned* __restrict__ mu, int ne, int n)
{
    const int i = blockIdx.x * blockDim.x + threadIdx.x;
    if (i >= ne + n) return;
    int s, d;
    if (i < ne) { s = src[i]; d = dst[i]; } else { s = d = i - ne; }
    float ev = s_src[s] + s_dst[d];
    ev = (ev > 0.0f) ? ev : 0.2f * ev;       // NEG_SLOPE = 0.2
    e[i] = ev;
    atomicMax(&mu[d], fkey(ev));
}

// ex = exp(e - max[dst]); denom[dst] += ex  (e overwritten in place)
__global__ void edge_exp(const int* __restrict__ dst,
                         float* __restrict__ e, const unsigned* __restrict__ mu,
                         float* __restrict__ denom, int ne, int n)
{
    const int i = blockIdx.x * blockDim.x + threadIdx.x;
    if (i >= ne + n) return;
    const int d = (i < ne) ? dst[i] : (i - ne);
    const float ex = __expf(e[i] - funkey(mu[d]));
    e[i] = ex;
    atomicAdd(&denom[d], ex);
}

// agg[dst] += xw[src] * (ex / denom[dst]); one wave per edge, float4 per lane
__global__ void edge_scatter(const int* __restrict__ src, const int* __restrict__ dst,
                             const float* __restrict__ e, const float* __restrict__ denom,
                             const float* __restrict__ xw, float* __restrict__ agg,
                             int ne, int n)
{
    const int w    = (blockIdx.x * blockDim.x + threadIdx.x) >> 5;
    const int lane = threadIdx.x & 31;
    if (w >= ne + n) return;
    int s, d;
    if (w < ne) { s = src[w]; d = dst[w]; } else { s = d = w - ne; }
    const float alpha = e[w] / denom[d];
    const float4 v = reinterpret_cast<const float4*>(xw + (size_t)s * HID)[lane];
    float* o = agg + (size_t)d * HID + lane * 4;
    atomicAdd(o + 0, v.x * alpha);
    atomicAdd(o + 1, v.y * alpha);
    atomicAdd(o + 2, v.z * alpha);
    atomicAdd(o + 3, v.w * alpha);
}

// io[i] = act(io[i] + bias[i%128]); optional mirror copy (e.g. into d_out hL)
__global__ void finalize_k(float* __restrict__ io, float* __restrict__ mirror,
                           const float* __restrict__ bias, int relu, int count)
{
    const int i = blockIdx.x * blockDim.x + threadIdx.x;
    if (i >= count) return;
    float v = io[i] + bias[i & (HID - 1)];
    if (relu) v = fmaxf(v, 0.0f);
    io[i] = v;
    if (mirror) mirror[i] = v;
}

// y[n] = sigmoid(<d2[n], Wh> + bh); one wave per node
__global__ void head_y(const float* __restrict__ d2, const float* __restrict__ Wh,
                       const float* __restrict__ bh, float* __restrict__ y, int n)
{
    const int w    = (blockIdx.x * blockDim.x + threadIdx.x) >> 5;
    const int lane = threadIdx.x & 31;
    if (w >= n) return;
    const float4 v  = reinterpret_cast<const float4*>(d2 + (size_t)w * HID)[lane];
    const float4 wh = reinterpret_cast<const float4*>(Wh)[lane];
    float p = v.x * wh.x + v.y * wh.y + v.z * wh.z + v.w * wh.w;
    for (int off = 16; off > 0; off >>= 1) p += __shfl_xor(p, off, 32);
    if (lane == 0) y[w] = sigmoidf_(p + bh[0]);
}

// column sums of hL into hbar[128] (grid-strided rows, one atomic per block/col)
__global__ void colsum(const float* __restrict__ hL, float* __restrict__ hbar, int n)
{
    const int c = threadIdx.x;          // 128 threads
    float s = 0.0f;
    for (int r = blockIdx.x; r < n; r += gridDim.x) s += hL[(size_t)r * HID + c];
    atomicAdd(&hbar[c], s);
}

// t = sigmoid(<colsum/n, Wt> + bt); single wave
__global__ void head_t(const float* __restrict__ hbar, const float* __restrict__ Wt,
                       const float* __restrict__ bt, float* __restrict__ t, int n)
{
    const int lane = threadIdx.x & 31;
    const float4 hb = reinterpret_cast<const float4*>(hbar)[lane];
    const float4 w  = reinterpret_cast<const float4*>(Wt)[lane];
    float p = hb.x * w.x + hb.y * w.y + hb.z * w.z + hb.w * w.w;
    for (int off = 16; off > 0; off >>= 1) p += __shfl_xor(p, off, 32);
    if (lane == 0) t[0] = sigmoidf_(p / (float)n + bt[0]);
}

extern "C" void kernel_launch(void* const* d_in, const int* in_sizes, int n_in,
                              void* d_out, int out_size, void* d_ws, size_t ws_size,
                              hipStream_t stream)
{
    const float* x    = (const float*)d_in[0];
    const float* h    = (const float*)d_in[1];
    const int*   ei   = (const int*)d_in[2];
    const float* Wenc = (const float*)d_in[3];
    const float* benc = (const float*)d_in[4];
    const float* Wg   = (const float*)d_in[5];
    const float* as_  = (const float*)d_in[6];
    const float* ad_  = (const float*)d_in[7];
    const float* bg   = (const float*)d_in[8];
    const float* Wd   = (const float*)d_in[9];
    const float* bd   = (const float*)d_in[10];
    const float* Wd1  = (const float*)d_in[11];
    const float* bd1  = (const float*)d_in[12];
    const float* Wh   = (const float*)d_in[13];
    const float* bh   = (const float*)d_in[14];
    const float* Wt   = (const float*)d_in[15];
    const float* bt   = (const float*)d_in[16];

    const int N  = in_sizes[1] / HID;      // 50000
    const int E  = in_sizes[2] / 2;        // 800000
    const int EA = E + N;                  // + self-loops
    const int NH = N * HID;
    const int* srcI = ei;
    const int* dstI = ei + E;

    float* ws   = (float*)d_ws;
    float* z    = ws;  ws += NH;
    float* A    = ws;  ws += NH;
    float* B    = ws;  ws += NH;
    float* xw   = ws;  ws += NH;
    float* ssrc = ws;  ws += N;
    float* sdst = ws;  ws += N;
    unsigned* mu = (unsigned*)ws; ws += N;   // mu & den contiguous -> one fill
    float* den  = ws;  ws += N;
    float* e    = ws;  ws += EA;
    float* hbar = ws;  ws += HID;

    float* yout = (float*)d_out;     // [N]
    float* tout = yout + N;          // [1]
    float* hLo  = tout + 1;          // [N*128] (element-wise writes only)

    const dim3 blk(256);
    const int gGemm  = (N + 127) / 128;
    const int gNodeW = (N * 32 + 255) / 256;
    const int gEdge  = (EA + 255) / 256;
    const int gEdgeW = (int)(((size_t)EA * 32 + 255) / 256);
    const int gElem  = (NH + 255) / 256;

    // ---- Encoder: z = concat(x, h) @ W_enc^T + b_enc (no activation)
    gemm_wmma<64, 128, 0><<<gGemm, blk, 0, stream>>>(x, h, Wenc, benc, z, N);

    auto conv = [&](const float* f, float* out, int layer, int relu, float* mirror) {
        const float* W  = Wg  + (size_t)layer * HID * HID;
        const float* as = as_ + (size_t)layer * HID;
        const float* ad = ad_ + (size_t)layer * HID;
        const float* bb = bg  + (size_t)layer * HID;
        gemm_wmma<128, 0, 0><<<gGemm, blk, 0, stream>>>(f, nullptr, W, nullptr, xw, N);
        node_scores<<<gNodeW, blk, 0, stream>>>(xw, as, ad, ssrc, sdst, N);
        fill_f32<<<gElem, blk, 0, stream>>>(out, 0.0f, NH);
        fill_f32<<<(2 * N + 255) / 256, blk, 0, stream>>>((float*)mu, 0.0f, 2 * N);
        edge_logits<<<gEdge, blk, 0, stream>>>(srcI, dstI, ssrc, sdst, e, mu, E, N);
        edge_exp<<<gEdge, blk, 0, stream>>>(dstI, e, mu, den, E, N);
        edge_scatter<<<gEdgeW, blk, 0, stream>>>(srcI, dstI, e, den, xw, out, E, N);
        finalize_k<<<gElem, blk, 0, stream>>>(out, mirror, bb, relu, NH);
    };

    // ---- GAT processor: relu(conv0), relu(conv1), relu(conv2), conv2 again
    conv(z, A, 0, 1, nullptr);
    conv(A, B, 1, 1, nullptr);
    conv(B, A, 2, 1, nullptr);
    conv(A, B, 2, 0, hLo);           // hL lives in B, mirrored into d_out

    // ---- Decoder
    gemm_wmma<128, 128, 1><<<gGemm, blk, 0, stream>>>(B, z, Wd, bd, A, N);        // d1
    gemm_wmma<128, 0, 1><<<gGemm, blk, 0, stream>>>(A, nullptr, Wd1, bd1, xw, N); // d2
    head_y<<<gNodeW, blk, 0, stream>>>(xw, Wh, bh, yout, N);

    // ---- Termination head
    fill_f32<<<1, 128, 0, stream>>>(hbar, 0.0f, HID);
    colsum<<<256, 128, 0, stream>>>(B, hbar, N);
    head_t<<<1, 32, 0, stream>>>(hbar, Wt, bt, tout, N);
}